// HERMERS_90726889161248
// MI455X (gfx1250) — compile-verified
//
#include <hip/hip_runtime.h>
#include <hip/hip_bf16.h>

// ============================================================================
// MI455X (gfx1250) implementation of the DeepDDS-like GNN forward pass.
//
// Dense GEMMs (tconv Q/K/V, cline MLP, hconv, rec bilinears, decoder) run on
// the CDNA5 matrix pipe via V_WMMA_F32_16X16X4_F32 (f32 precision == ref).
// One wave32 computes one 16x16 output tile; 4 waves/block cover 4 adjacent
// N-tiles for A-tile reuse in L1. Inner loop: unconditional b64 loads,
// 2x unrolled, dual accumulators; ragged K handled by one masked tail WMMA.
//
// Input leaf order (setup_inputs() dict insertion order, params expanded):
//   0 drug_x(60000x78) 1 cline_x(128x954) 2 hyperedge_weight(30000)
//   3..20  drug_conv: 3 layers x {Wq,bq,Wk,bk,Wv,bv}
//   21..26 drug_bn: (g,b) x3
//   27 cline_first_W(954x256) 28 cline_first_b
//   29,30 cline_W[2] 31,32 cline_b[2]
//   33..38 hg_bn (g,b)x3   39..41 hg_W   42..44 hg_b
//   45..47 gate_W(256x1)   48..50 gate_b(1)
//   51 drug_rec_W 52 cline_rec_W
//   53 dec_W1(768x384) 54 dec_b1 55 dec_W2(384x192) 56 dec_b2 57 dec_W3(192x1) 58 dec_b3
//   59 drug_adj(2x120000) 60 ibatch(60000) 61 H(2x90000)
//   62 druga_id 63 drugb_id 64 cline_id
// Output: pred(16384) | rec_drug(2000x2000) | rec_cline(128x128), f32.
// Workspace: ~314 MB of f32 scratch (5 node-sized regions + small tail).
// ============================================================================

#define NA    60000
#define ECNT  120000
#define NDRUG 2000
#define NCL   128
#define NTOT  2128
#define NEH   30000
#define HE3   90000
#define DD    256
#define NB    16384

typedef __attribute__((ext_vector_type(2))) float v2f;
typedef __attribute__((ext_vector_type(8))) float v8f;

// ------------------------------- utility kernels ----------------------------

__global__ void k_fill(float* __restrict__ p, float v, long n) {
  long i = (long)blockIdx.x * blockDim.x + threadIdx.x;
  long st = (long)gridDim.x * blockDim.x;
  for (; i < n; i += st) p[i] = v;
}

__global__ void k_copy(float* __restrict__ d, const float* __restrict__ s, long n) {
  long i = (long)blockIdx.x * blockDim.x + threadIdx.x;
  long st = (long)gridDim.x * blockDim.x;
  for (; i < n; i += st) d[i] = s[i];
}

__device__ __forceinline__ float atomicMaxF(float* addr, float val) {
  int* ai = (int*)addr;
  int old = __float_as_int(*addr);
  while (__int_as_float(old) < val) {
    int assumed = old;
    old = atomicCAS(ai, assumed, __float_as_int(val));
    if (old == assumed) break;
  }
  return __int_as_float(old);
}

// ------------------------------- WMMA GEMM ----------------------------------
// C[M,N] = epilogue( A[M,K] @ op(B) + bias ), op(B) = B[K,N] or B[N,K]^T.
// Requires: M % 16 == 0, N % 16 == 0 (true at all call sites); K arbitrary.
// lda/ldb even (8B-aligned v2f loads) at all call sites.
// mode: 0=none 1=relu 2=tanh 3=sigmoid 4=resid+relu
// Block = 128 threads = 4 waves; wave w handles N-tile (blockIdx.y*4 + w).
__global__ void k_gemm(const float* __restrict__ A, int lda,
                       const float* __restrict__ B, int ldb, int transB,
                       const float* __restrict__ bias,
                       const float* __restrict__ resid, int ldr,
                       float* __restrict__ C, int ldc,
                       int M, int N, int K, int mode)
{
  const int lane  = threadIdx.x & 31;        // lane within wave32
  const int wave  = threadIdx.x >> 5;        // 0..3
  const int nTile = blockIdx.y * 4 + wave;
  if (nTile * 16 >= N) return;               // wave-uniform exit

  const int r16   = lane & 15;
  const int khalf = (lane >> 4) << 1;        // lanes 0-15 -> K {0,1}; 16-31 -> {2,3}
  const long rowA = (long)blockIdx.x * 16 + r16;
  const long col  = (long)nTile * 16 + r16;

  const float* ap = A + rowA * lda + khalf;  // lane's A stream: A[row][kb+khalf..+1]
  v8f acc0 = {0.f,0.f,0.f,0.f,0.f,0.f,0.f,0.f};
  v8f acc1 = {0.f,0.f,0.f,0.f,0.f,0.f,0.f,0.f};

  const int K4 = K & ~3;
  int kb = 0;
  if (transB) {
    const float* bp = B + col * ldb + khalf; // B^T: contiguous along K
    for (; kb + 8 <= K4; kb += 8) {
      v2f a0 = *(const v2f*)(ap + kb);
      v2f b0 = *(const v2f*)(bp + kb);
      v2f a1 = *(const v2f*)(ap + kb + 4);
      v2f b1 = *(const v2f*)(bp + kb + 4);
      acc0 = __builtin_amdgcn_wmma_f32_16x16x4_f32(false, a0, false, b0, (short)0, acc0, false, false);
      acc1 = __builtin_amdgcn_wmma_f32_16x16x4_f32(false, a1, false, b1, (short)0, acc1, false, false);
    }
    for (; kb < K4; kb += 4) {
      v2f a0 = *(const v2f*)(ap + kb);
      v2f b0 = *(const v2f*)(bp + kb);
      acc0 = __builtin_amdgcn_wmma_f32_16x16x4_f32(false, a0, false, b0, (short)0, acc0, false, false);
    }
  } else {
    const float* bp = B + (long)khalf * ldb + col; // B[kb+khalf][col]
    const long ld2 = (long)ldb;
    for (; kb + 8 <= K4; kb += 8) {
      v2f a0 = *(const v2f*)(ap + kb);
      v2f a1 = *(const v2f*)(ap + kb + 4);
      const float* b = bp + (long)kb * ld2;
      v2f b0 = { b[0], b[ld2] };
      v2f b1 = { b[4 * ld2], b[5 * ld2] };
      acc0 = __builtin_amdgcn_wmma_f32_16x16x4_f32(false, a0, false, b0, (short)0, acc0, false, false);
      acc1 = __builtin_amdgcn_wmma_f32_16x16x4_f32(false, a1, false, b1, (short)0, acc1, false, false);
    }
    for (; kb < K4; kb += 4) {
      v2f a0 = *(const v2f*)(ap + kb);
      const float* b = bp + (long)kb * ld2;
      v2f b0 = { b[0], b[ld2] };
      acc0 = __builtin_amdgcn_wmma_f32_16x16x4_f32(false, a0, false, b0, (short)0, acc0, false, false);
    }
  }
  if (K4 < K) {                               // ragged K tail (K=78, 954): masked
    int k0 = K4 + khalf, k1 = k0 + 1;
    int k0c = (k0 < K) ? k0 : (K - 1);
    int k1c = (k1 < K) ? k1 : (K - 1);
    float m0 = (k0 < K) ? 1.f : 0.f;
    float m1 = (k1 < K) ? 1.f : 0.f;
    v2f a0 = { A[rowA * lda + k0c] * m0, A[rowA * lda + k1c] * m1 };
    v2f b0;
    if (transB) b0 = (v2f){ B[col * ldb + k0c] * m0, B[col * ldb + k1c] * m1 };
    else        b0 = (v2f){ B[(long)k0c * ldb + col] * m0, B[(long)k1c * ldb + col] * m1 };
    acc0 = __builtin_amdgcn_wmma_f32_16x16x4_f32(false, a0, false, b0, (short)0, acc0, false, false);
  }

  v8f acc;
#pragma unroll
  for (int j = 0; j < 8; ++j) acc[j] = acc0[j] + acc1[j];

  float bb = bias ? bias[col] : 0.f;
  long rbase = (long)blockIdx.x * 16 + ((lane < 16) ? 0 : 8);
#pragma unroll
  for (int j = 0; j < 8; ++j) {
    long r = rbase + j;
    float v = acc[j] + bb;
    if      (mode == 1) v = fmaxf(v, 0.f);
    else if (mode == 2) v = tanhf(v);
    else if (mode == 3) v = 1.f / (1.f + expf(-v));
    else if (mode == 4) v = resid[r * ldr + col] + fmaxf(v, 0.f);
    C[r * ldc + col] = v;
  }
}

// ---------------------------- transformer conv ------------------------------

__global__ void k_edge_score(const float* __restrict__ q, const float* __restrict__ k,
                             const int* __restrict__ src, const int* __restrict__ dst,
                             float* __restrict__ s, int nE) {
  int i = blockIdx.x * blockDim.x + threadIdx.x;
  if (i >= nE * 4) return;
  int e = i >> 2, h = i & 3;
  const float* qp = q + (long)dst[e] * DD + h * 64;
  const float* kp = k + (long)src[e] * DD + h * 64;
  float acc = 0.f;
#pragma unroll 8
  for (int j = 0; j < 64; ++j) acc += qp[j] * kp[j];
  s[i] = acc * 0.125f;   // / sqrt(64)
}

__global__ void k_seg_max(const float* __restrict__ s, const int* __restrict__ dst,
                          float* __restrict__ smax, int nE) {
  int i = blockIdx.x * blockDim.x + threadIdx.x;
  if (i >= nE * 4) return;
  int e = i >> 2, h = i & 3;
  atomicMaxF(&smax[(long)dst[e] * 4 + h], s[i]);
}

__global__ void k_edge_exp(float* __restrict__ s, const float* __restrict__ smax,
                           const int* __restrict__ dst, float* __restrict__ den, int nE) {
  int i = blockIdx.x * blockDim.x + threadIdx.x;
  if (i >= nE * 4) return;
  int e = i >> 2, h = i & 3;
  float ex = expf(s[i] - smax[(long)dst[e] * 4 + h]);
  s[i] = ex;
  atomicAdd(&den[(long)dst[e] * 4 + h], ex);
}

__global__ void k_edge_aggr(const float* __restrict__ ex, const float* __restrict__ den,
                            const float* __restrict__ v, const int* __restrict__ src,
                            const int* __restrict__ dst, float* __restrict__ out, long total) {
  long i = (long)blockIdx.x * blockDim.x + threadIdx.x;
  long st = (long)gridDim.x * blockDim.x;
  for (; i < total; i += st) {
    int e = (int)(i >> 8), d = (int)(i & 255), h = d >> 6;
    float dn = den[(long)dst[e] * 4 + h];
    float alpha = ex[(long)e * 4 + h] / dn;
    atomicAdd(&out[(long)dst[e] * DD + d], alpha * v[(long)src[e] * DD + d]);
  }
}

// ---------------------------------- batchnorm -------------------------------

__global__ void k_bn_stats(const float* __restrict__ x, float* __restrict__ mean,
                           float* __restrict__ var, int M, int relu_flag) {
  int c = blockIdx.x;  // one block per column, D=256
  __shared__ float ssum[256], ssq[256];
  float s = 0.f, sq = 0.f;
  for (int r = threadIdx.x; r < M; r += blockDim.x) {
    float v = x[(long)r * DD + c];
    if (relu_flag) v = fmaxf(v, 0.f);
    s += v; sq += v * v;
  }
  ssum[threadIdx.x] = s; ssq[threadIdx.x] = sq;
  __syncthreads();
  for (int o = blockDim.x >> 1; o > 0; o >>= 1) {
    if (threadIdx.x < o) { ssum[threadIdx.x] += ssum[threadIdx.x + o];
                           ssq[threadIdx.x]  += ssq[threadIdx.x + o]; }
    __syncthreads();
  }
  if (threadIdx.x == 0) {
    float mu = ssum[0] / (float)M;
    mean[c] = mu;
    var[c]  = ssq[0] / (float)M - mu * mu;
  }
}

__global__ void k_bn_apply(const float* __restrict__ x, const float* __restrict__ mean,
                           const float* __restrict__ var, const float* __restrict__ gw,
                           const float* __restrict__ gb, const float* __restrict__ resid,
                           float* __restrict__ out, long total, int relu_flag) {
  long i = (long)blockIdx.x * blockDim.x + threadIdx.x;
  long st = (long)gridDim.x * blockDim.x;
  for (; i < total; i += st) {
    int c = (int)(i & 255);
    float v = x[i];
    if (relu_flag) v = fmaxf(v, 0.f);
    float y = (v - mean[c]) * rsqrtf(var[c] + 1e-5f) * gw[c] + gb[c];
    out[i] = resid ? (resid[i] + y) : y;
  }
}

// ----------------------------------- pooling --------------------------------

__global__ void k_pool(const float* __restrict__ h, const int* __restrict__ ib,
                       float* __restrict__ sum, float* __restrict__ cnt, long total) {
  long i = (long)blockIdx.x * blockDim.x + threadIdx.x;
  long st = (long)gridDim.x * blockDim.x;
  for (; i < total; i += st) {
    int a = (int)(i >> 8), d = (int)(i & 255);
    atomicAdd(&sum[(long)ib[a] * DD + d], h[i]);
    if (d == 0) atomicAdd(&cnt[ib[a]], 1.f);
  }
}

__global__ void k_pool_div(float* __restrict__ sum, const float* __restrict__ cnt, long total) {
  long i = (long)blockIdx.x * blockDim.x + threadIdx.x;
  if (i >= total) return;
  sum[i] /= fmaxf(cnt[i >> 8], 1.f);
}

// -------------------------------- hypergraph --------------------------------

__global__ void k_bdeg(const int* __restrict__ edge, float* __restrict__ bdeg, int n) {
  int i = blockIdx.x * blockDim.x + threadIdx.x;
  if (i < n) atomicAdd(&bdeg[edge[i]], 1.f);
}
__global__ void k_ddeg(const int* __restrict__ node, const int* __restrict__ edge,
                       const float* __restrict__ w, float* __restrict__ ddeg, int n) {
  int i = blockIdx.x * blockDim.x + threadIdx.x;
  if (i < n) atomicAdd(&ddeg[node[i]], w[edge[i]]);
}
__global__ void k_inv(float* __restrict__ x, int n) {
  int i = blockIdx.x * blockDim.x + threadIdx.x;
  if (i < n) x[i] = (x[i] > 0.f) ? (1.f / x[i]) : 0.f;
}

__global__ void k_gate(const float* __restrict__ X, const float* __restrict__ gw,
                       const float* __restrict__ gb, float* __restrict__ gate, int M) {
  int r = blockIdx.x * blockDim.x + threadIdx.x;
  if (r >= M) return;
  float acc = gb[0];
  for (int c = 0; c < DD; ++c) acc += X[(long)r * DD + c] * gw[c];
  gate[r] = 1.f / (1.f + expf(-acc));
}

__global__ void k_scatter_edge(const float* __restrict__ XW, const int* __restrict__ node,
                               const int* __restrict__ edge, float* __restrict__ me, long total) {
  long i = (long)blockIdx.x * blockDim.x + threadIdx.x;
  long st = (long)gridDim.x * blockDim.x;
  for (; i < total; i += st) {
    int t = (int)(i >> 8), d = (int)(i & 255);
    atomicAdd(&me[(long)edge[t] * DD + d], XW[(long)node[t] * DD + d]);
  }
}
__global__ void k_scale_rows(float* __restrict__ me, const float* __restrict__ inv, long total) {
  long i = (long)blockIdx.x * blockDim.x + threadIdx.x;
  long st = (long)gridDim.x * blockDim.x;
  for (; i < total; i += st) me[i] *= inv[i >> 8];
}
__global__ void k_scatter_node(const float* __restrict__ me, const int* __restrict__ node,
                               const int* __restrict__ edge, const float* __restrict__ w,
                               float* __restrict__ hout, long total) {
  long i = (long)blockIdx.x * blockDim.x + threadIdx.x;
  long st = (long)gridDim.x * blockDim.x;
  for (; i < total; i += st) {
    int t = (int)(i >> 8), d = (int)(i & 255);
    atomicAdd(&hout[(long)node[t] * DD + d], w[edge[t]] * me[(long)edge[t] * DD + d]);
  }
}
__global__ void k_hg_update(float* __restrict__ X, const float* __restrict__ hout,
                            const float* __restrict__ dinv, const float* __restrict__ hb,
                            const float* __restrict__ gate, long total) {
  long i = (long)blockIdx.x * blockDim.x + threadIdx.x;
  long st = (long)gridDim.x * blockDim.x;
  for (; i < total; i += st) {
    int r = (int)(i >> 8), c = (int)(i & 255);
    float v = hout[i] * dinv[r] + hb[c];
    X[i] += fmaxf(v, 0.f) * gate[r];
  }
}
__global__ void k_combine(const float* __restrict__ X, const float* __restrict__ X0,
                          float* __restrict__ g, long n) {
  long i = (long)blockIdx.x * blockDim.x + threadIdx.x;
  long st = (long)gridDim.x * blockDim.x;
  for (; i < n; i += st) g[i] = 2.f * X[i] + X0[i];
}

// ---------------------------------- decoder ---------------------------------

__global__ void k_cand(const float* __restrict__ g, const int* __restrict__ ia,
                       const int* __restrict__ ib, const int* __restrict__ ic,
                       float* __restrict__ cand, long total) {
  long i = (long)blockIdx.x * blockDim.x + threadIdx.x;
  long st = (long)gridDim.x * blockDim.x;
  for (; i < total; i += st) {
    int b = (int)(i / 768), j = (int)(i % 768);
    int which = j >> 8, c = j & 255;
    int id = (which == 0) ? ia[b] : ((which == 1) ? ib[b] : ic[b]);
    cand[i] = g[(long)id * DD + c];
  }
}

__global__ void k_dec3(const float* __restrict__ hd, const float* __restrict__ W3,
                       const float* __restrict__ b3, float* __restrict__ pred, int M, int K) {
  int r = blockIdx.x * blockDim.x + threadIdx.x;
  if (r >= M) return;
  float acc = b3[0];
  for (int k = 0; k < K; ++k) acc += hd[(long)r * K + k] * W3[k];
  pred[r] = 1.f / (1.f + expf(-acc));
}

// --------------------------------- host side --------------------------------

static inline int cdiv_i(long a, long b) { return (int)((a + b - 1) / b); }

static inline void launch_gemm(hipStream_t st, const float* A, int lda,
                               const float* B, int ldb, int transB,
                               const float* bias, const float* resid, int ldr,
                               float* C, int ldc, int M, int N, int K, int mode) {
  // 4 waves per block -> 4 adjacent N-tiles
  dim3 grid(cdiv_i(M, 16), cdiv_i(cdiv_i(N, 16), 4));
  k_gemm<<<grid, 128, 0, st>>>(A, lda, B, ldb, transB, bias, resid, ldr, C, ldc, M, N, K, mode);
}

extern "C" void kernel_launch(void* const* d_in, const int* in_sizes, int n_in,
                              void* d_out, int out_size, void* d_ws, size_t ws_size,
                              hipStream_t stream) {
  const float* drug_x   = (const float*)d_in[0];
  const float* cline_x  = (const float*)d_in[1];
  const float* he_w     = (const float*)d_in[2];
  auto P = [&](int i) { return (const float*)d_in[i]; };
  const int* drug_adj = (const int*)d_in[59];
  const int* ibatch   = (const int*)d_in[60];
  const int* Hmat     = (const int*)d_in[61];
  const int* druga    = (const int*)d_in[62];
  const int* drugb    = (const int*)d_in[63];
  const int* clineid  = (const int*)d_in[64];
  float* out = (float*)d_out;

  const int* esrc = drug_adj;            // drug_adj[0]
  const int* edst = drug_adj + ECNT;     // drug_adj[1]
  const int* hnode = Hmat;               // H[0]
  const int* hedge = Hmat + HE3;         // H[1]

  // ------------------ workspace layout (floats) ------------------
  const long SZ = (long)NA * DD;         // 15,360,000
  float* W0 = (float*)d_ws;
  float* bufH = W0;                       // node features (stage A)
  float* bufQ = W0 + SZ;                  // Q / later hypergraph region
  float* bufK = W0 + 2 * SZ;              // K / later cand
  float* bufV = W0 + 3 * SZ;              // V / later hd1, hd2
  float* bufO = W0 + 4 * SZ;              // tconv out
  float* tail = W0 + 5 * SZ;              // small persistent tail
  float* sEx   = tail;                    // E*4
  float* smax  = tail + 480000;           // NA*4
  float* den   = tail + 720000;           // NA*4
  float* cnt   = tail + 960000;           // NDRUG
  float* dEmb  = tail + 962000;           // NDRUG*DD
  float* cA    = tail + 1474000;          // 128*256
  float* cB    = tail + 1506768;          // 128*256
  float* meanb = tail + 1539536;          // 256
  float* varb  = tail + 1539792;          // 256
  // hypergraph overlays in bufQ region:
  float* Xm   = bufQ;                     // NTOT*DD
  float* X0   = bufQ + 544768;
  float* bnX  = bufQ + 1089536;
  float* XW   = bufQ + 1634304;
  float* hout = bufQ + 2179072;
  float* gate = bufQ + 2723840;           // NTOT
  float* bdeg = bufQ + 2726000;           // NEH
  float* ddeg = bufQ + 2756000;           // NTOT
  float* me   = bufQ + 2760000;           // NEH*DD
  float* Tbuf = bufQ + 10440000;          // NDRUG*DD
  float* gbuf = bufQ + 10952000;          // NTOT*DD
  float* cand = bufK;                     // NB*768
  float* hd1  = bufV;                     // NB*384
  float* hd2  = bufV + (long)NB * 384;    // NB*192

  const int T = 256;
  auto gs = [&](long n) { int b = cdiv_i(n, T); return (b > 262144) ? 262144 : b; };

  // ================= Stage A: 3 transformer-conv layers =================
  const float* xin = drug_x; int Kin = 78, ldx = 78;
  for (int layer = 0; layer < 3; ++layer) {
    int pb = 3 + layer * 6;
    launch_gemm(stream, xin, ldx, P(pb + 0), DD, 0, P(pb + 1), nullptr, 0, bufQ, DD, NA, DD, Kin, 0);
    launch_gemm(stream, xin, ldx, P(pb + 2), DD, 0, P(pb + 3), nullptr, 0, bufK, DD, NA, DD, Kin, 0);
    launch_gemm(stream, xin, ldx, P(pb + 4), DD, 0, P(pb + 5), nullptr, 0, bufV, DD, NA, DD, Kin, 0);

    k_fill<<<gs((long)NA * 4), T, 0, stream>>>(smax, -3.0e38f, (long)NA * 4);
    k_fill<<<gs((long)NA * 4), T, 0, stream>>>(den, 0.f, (long)NA * 4);
    k_edge_score<<<cdiv_i((long)ECNT * 4, T), T, 0, stream>>>(bufQ, bufK, esrc, edst, sEx, ECNT);
    k_seg_max<<<cdiv_i((long)ECNT * 4, T), T, 0, stream>>>(sEx, edst, smax, ECNT);
    k_edge_exp<<<cdiv_i((long)ECNT * 4, T), T, 0, stream>>>(sEx, smax, edst, den, ECNT);
    k_fill<<<gs(SZ), T, 0, stream>>>(bufO, 0.f, SZ);
    k_edge_aggr<<<gs((long)ECNT * DD), T, 0, stream>>>(sEx, den, bufV, esrc, edst, bufO, (long)ECNT * DD);

    // h = [h +] bn(relu(out))
    k_bn_stats<<<DD, T, 0, stream>>>(bufO, meanb, varb, NA, 1);
    const float* resid = (layer == 0) ? nullptr : bufH;
    k_bn_apply<<<gs(SZ), T, 0, stream>>>(bufO, meanb, varb, P(21 + layer * 2), P(22 + layer * 2),
                                         resid, bufH, SZ, 1);
    xin = bufH; Kin = DD; ldx = DD;
  }

  // ================= Stage B: mean pool atoms -> drugs =================
  k_fill<<<gs(NDRUG), T, 0, stream>>>(cnt, 0.f, NDRUG);
  k_fill<<<gs((long)NDRUG * DD), T, 0, stream>>>(dEmb, 0.f, (long)NDRUG * DD);
  k_pool<<<gs(SZ), T, 0, stream>>>(bufH, ibatch, dEmb, cnt, SZ);
  k_pool_div<<<cdiv_i((long)NDRUG * DD, T), T, 0, stream>>>(dEmb, cnt, (long)NDRUG * DD);

  // ================= Stage C: cell-line MLP =================
  launch_gemm(stream, cline_x, 954, P(27), DD, 0, P(28), nullptr, 0, cA, DD, NCL, DD, 954, 2);
  for (int i = 0; i < 2; ++i) {
    launch_gemm(stream, cA, DD, P(29 + i), DD, 0, P(31 + i), cA, DD, cB, DD, NCL, DD, DD, 4);
    float* t = cA; cA = cB; cB = t;
  }

  // ================= Stage D: assemble X, degrees =================
  k_copy<<<gs((long)NDRUG * DD), T, 0, stream>>>(Xm, dEmb, (long)NDRUG * DD);
  k_copy<<<gs((long)NCL * DD), T, 0, stream>>>(Xm + (long)NDRUG * DD, cA, (long)NCL * DD);
  k_copy<<<gs((long)NTOT * DD), T, 0, stream>>>(X0, Xm, (long)NTOT * DD);

  k_fill<<<gs(NEH), T, 0, stream>>>(bdeg, 0.f, NEH);
  k_fill<<<gs(NTOT), T, 0, stream>>>(ddeg, 0.f, NTOT);
  k_bdeg<<<cdiv_i(HE3, T), T, 0, stream>>>(hedge, bdeg, HE3);
  k_ddeg<<<cdiv_i(HE3, T), T, 0, stream>>>(hnode, hedge, he_w, ddeg, HE3);
  k_inv<<<cdiv_i(NEH, T), T, 0, stream>>>(bdeg, NEH);
  k_inv<<<cdiv_i(NTOT, T), T, 0, stream>>>(ddeg, NTOT);

  // ================= Stage E: 3 gated hypergraph conv layers =================
  const long XT = (long)NTOT * DD;
  for (int i = 0; i < 3; ++i) {
    k_gate<<<cdiv_i(NTOT, T), T, 0, stream>>>(Xm, P(45 + i), P(48 + i), gate, NTOT);
    k_bn_stats<<<DD, T, 0, stream>>>(Xm, meanb, varb, NTOT, 0);
    k_bn_apply<<<gs(XT), T, 0, stream>>>(Xm, meanb, varb, P(33 + 2 * i), P(34 + 2 * i),
                                         nullptr, bnX, XT, 0);
    launch_gemm(stream, bnX, DD, P(39 + i), DD, 0, nullptr, nullptr, 0, XW, DD, NTOT, DD, DD, 0);
    k_fill<<<gs((long)NEH * DD), T, 0, stream>>>(me, 0.f, (long)NEH * DD);
    k_scatter_edge<<<gs((long)HE3 * DD), T, 0, stream>>>(XW, hnode, hedge, me, (long)HE3 * DD);
    k_scale_rows<<<gs((long)NEH * DD), T, 0, stream>>>(me, bdeg, (long)NEH * DD);
    k_fill<<<gs(XT), T, 0, stream>>>(hout, 0.f, XT);
    k_scatter_node<<<gs((long)HE3 * DD), T, 0, stream>>>(me, hnode, hedge, he_w, hout, (long)HE3 * DD);
    k_hg_update<<<gs(XT), T, 0, stream>>>(Xm, hout, ddeg, P(42 + i), gate, XT);
  }
  k_combine<<<gs(XT), T, 0, stream>>>(Xm, X0, gbuf, XT);

  // ================= Stage F: reconstruction bilinears =================
  float* rec_drug  = out + NB;
  float* rec_cline = out + NB + (long)NDRUG * NDRUG;
  launch_gemm(stream, gbuf, DD, P(51), DD, 0, nullptr, nullptr, 0, Tbuf, DD, NDRUG, DD, DD, 0);
  launch_gemm(stream, Tbuf, DD, gbuf, DD, 1, nullptr, nullptr, 0, rec_drug, NDRUG, NDRUG, NDRUG, DD, 3);
  launch_gemm(stream, gbuf + (long)NDRUG * DD, DD, P(52), DD, 0, nullptr, nullptr, 0, Tbuf, DD, NCL, DD, DD, 0);
  launch_gemm(stream, Tbuf, DD, gbuf + (long)NDRUG * DD, DD, 1, nullptr, nullptr, 0, rec_cline, NCL, NCL, NCL, DD, 3);

  // ================= Stage G: decoder MLP =================
  k_cand<<<gs((long)NB * 768), T, 0, stream>>>(gbuf, druga, drugb, clineid, cand, (long)NB * 768);
  launch_gemm(stream, cand, 768, P(53), 384, 0, P(54), nullptr, 0, hd1, 384, NB, 384, 768, 1);
  launch_gemm(stream, hd1, 384, P(55), 192, 0, P(56), nullptr, 0, hd2, 192, NB, 192, 384, 1);
  k_dec3<<<cdiv_i(NB, T), T, 0, stream>>>(hd2, P(57), P(58), out, NB, 192);
}